// GraphEncoder_20641612824813
// MI455X (gfx1250) — compile-verified
//
#include <hip/hip_runtime.h>
#include <hip/hip_bf16.h>
#include <math.h>

// ---------------------------------------------------------------------------
// GAT encoder for MI455X (gfx1250, wave32).
//   Layer1: GAT(128 -> 4 heads x 64, concat) ; Layer2: GAT(256 -> 64, 1 head)
//   Pool:   gate MLP + global softmax + weighted sum -> [1,64]
// All dense GEMMs (X@W1, X@W2, x2@Wg1) use V_WMMA_F32_16X16X4_F32 (fp32 WMMA,
// full reference precision). Edge phase uses f32 global atomics.
// ---------------------------------------------------------------------------

typedef __attribute__((ext_vector_type(2))) float v2f;
typedef __attribute__((ext_vector_type(8))) float v8f;

#define LEAKY_SLOPE 0.2f
#define SM_EPS 1e-16f

// ---------------- float atomic max (sign-split trick -> native int atomics) --
__device__ __forceinline__ void atomicMaxF(float* addr, float val) {
    if (val >= 0.0f) atomicMax((int*)addr, __float_as_int(val));
    else             atomicMin((unsigned int*)addr, __float_as_uint(val));
}

// ---------------- generic fill ----------------------------------------------
__global__ void fill_k(float* __restrict__ p, long long n, float v) {
    long long i = (long long)blockIdx.x * blockDim.x + threadIdx.x;
    if (i < n) p[i] = v;
}

// ---------------- fp32 WMMA GEMM: OUT[N,COLS] = X[N,K] @ W[K,COLS] ----------
// grid.x covers row tiles (4 waves per block, one 16-row tile per wave)
// grid.y = COLS/16 ; the 16-column slab of W is staged in LDS per block.
__global__ __launch_bounds__(128)
void gemm_wmma_f32(const float* __restrict__ X, const float* __restrict__ W,
                   float* __restrict__ OUT, int Nrows, int K, int COLS) {
    __shared__ float sW[4096];                    // up to K=256 x 16 cols (16 KB)
    const int col0 = blockIdx.y * 16;
    for (int idx = threadIdx.x; idx < K * 16; idx += blockDim.x) {
        int k = idx >> 4, c = idx & 15;
        sW[idx] = W[(size_t)k * COLS + col0 + c];
    }
    __syncthreads();

    const int wave = threadIdx.x >> 5;            // 0..3
    const int lane = threadIdx.x & 31;
    const int row0 = (blockIdx.x * 4 + wave) * 16;
    if (row0 >= Nrows) return;                    // wave-uniform exit

    const int m    = lane & 15;                   // M index within tile
    const int koff = (lane >> 4) * 2;             // lane-half selects K+0 / K+2
    int arow = row0 + m; if (arow >= Nrows) arow = Nrows - 1;   // clamped read
    const float* __restrict__ xrow = X + (size_t)arow * K;

    v8f acc = {0.f, 0.f, 0.f, 0.f, 0.f, 0.f, 0.f, 0.f};
    for (int k0 = 0; k0 < K; k0 += 4) {
        v2f a, b;
        a.x = xrow[k0 + koff + 0];
        a.y = xrow[k0 + koff + 1];
        b.x = sW[(k0 + koff + 0) * 16 + m];
        b.y = sW[(k0 + koff + 1) * 16 + m];
        // 8-arg form: (neg_a, A, neg_b, B, c_mod, C, reuse_a, reuse_b)
        acc = __builtin_amdgcn_wmma_f32_16x16x4_f32(
                false, a, false, b, (short)0, acc, false, false);
    }
    // C/D layout: VGPR v holds row (row0 + v + 8*(lane>=16)), col = col0 + m
    const int rbase = row0 + ((lane >> 4) << 3);
    #pragma unroll
    for (int v = 0; v < 8; ++v) {
        int r = rbase + v;
        if (r < Nrows) OUT[(size_t)r * COLS + col0 + m] = acc[v];
    }
}

// ---------------- per-node attention coefficients ---------------------------
__global__ void attn_coef_k(const float* __restrict__ xp,
                            const float* __restrict__ a_src,
                            const float* __restrict__ a_dst,
                            float* __restrict__ asrc, float* __restrict__ adst,
                            int Nn, int H, int C) {
    int idx = blockIdx.x * blockDim.x + threadIdx.x;
    if (idx >= Nn * H) return;
    int n = idx / H, h = idx - n * H;
    const float* v  = xp + ((size_t)n * H + h) * C;
    const float* as = a_src + h * C;
    const float* ad = a_dst + h * C;
    float s0 = 0.f, s1 = 0.f;
    for (int c = 0; c < C; ++c) { float x = v[c]; s0 += x * as[c]; s1 += x * ad[c]; }
    asrc[idx] = s0; adst[idx] = s1;
}

__device__ __forceinline__ void edge_nodes(const long long* __restrict__ srcA,
                                           const long long* __restrict__ dstA,
                                           int E, int e, int& s, int& d) {
    if (e < E) { s = (int)srcA[e]; d = (int)dstA[e]; }
    else       { s = d = e - E; }                  // self-loop
}

__device__ __forceinline__ float leaky_e(const float* asrc, const float* adst,
                                         int s, int d, int h, int H) {
    float v = asrc[s * H + h] + adst[d * H + h];
    return v > 0.f ? v : LEAKY_SLOPE * v;
}

// ---------------- segment max over dst --------------------------------------
__global__ void edge_max_k(const long long* __restrict__ srcA,
                           const long long* __restrict__ dstA, int E, int Esl,
                           const float* __restrict__ asrc,
                           const float* __restrict__ adst,
                           float* __restrict__ mseg, int H) {
    int idx = blockIdx.x * blockDim.x + threadIdx.x;
    if (idx >= Esl * H) return;
    int e = idx / H, h = idx - e * H;
    int s, d; edge_nodes(srcA, dstA, E, e, s, d);
    atomicMaxF(&mseg[d * H + h], leaky_e(asrc, adst, s, d, h, H));
}

// ---------------- segment sum of exp ----------------------------------------
__global__ void edge_sum_k(const long long* __restrict__ srcA,
                           const long long* __restrict__ dstA, int E, int Esl,
                           const float* __restrict__ asrc,
                           const float* __restrict__ adst,
                           const float* __restrict__ mseg,
                           float* __restrict__ denom, int H) {
    int idx = blockIdx.x * blockDim.x + threadIdx.x;
    if (idx >= Esl * H) return;
    int e = idx / H, h = idx - e * H;
    int s, d; edge_nodes(srcA, dstA, E, e, s, d);
    float v = leaky_e(asrc, adst, s, d, h, H);
    atomicAdd(&denom[d * H + h], expf(v - mseg[d * H + h]));
}

// ---------------- weighted message scatter ----------------------------------
__global__ void edge_scatter_k(const long long* __restrict__ srcA,
                               const long long* __restrict__ dstA, int E, int Esl,
                               const float* __restrict__ asrc,
                               const float* __restrict__ adst,
                               const float* __restrict__ mseg,
                               const float* __restrict__ denom,
                               const float* __restrict__ xp,
                               float* __restrict__ out, int H, int C) {
    int idx = blockIdx.x * blockDim.x + threadIdx.x;
    if (idx >= Esl * H) return;
    int e = idx / H, h = idx - e * H;
    int s, d; edge_nodes(srcA, dstA, E, e, s, d);
    float v = leaky_e(asrc, adst, s, d, h, H);
    float alpha = expf(v - mseg[d * H + h]) / (denom[d * H + h] + SM_EPS);
    const float* xs = xp + ((size_t)s * H + h) * C;
    float*       od = out + ((size_t)d * H + h) * C;
    for (int c = 0; c < C; c += 4) {
        float4 x4 = *(const float4*)(xs + c);
        atomicAdd(od + c + 0, x4.x * alpha);
        atomicAdd(od + c + 1, x4.y * alpha);
        atomicAdd(od + c + 2, x4.z * alpha);
        atomicAdd(od + c + 3, x4.w * alpha);
    }
}

// ---------------- bias + ELU (in place) -------------------------------------
__global__ void elu_bias_k(float* __restrict__ x, const float* __restrict__ b,
                           long long total, int F) {
    long long idx = (long long)blockIdx.x * blockDim.x + threadIdx.x;
    if (idx >= total) return;
    float v = x[idx] + b[idx % F];
    x[idx] = v > 0.f ? v : (expf(v) - 1.f);
}

// ---------------- bias + ReLU (in place) ------------------------------------
__global__ void relu_bias_k(float* __restrict__ x, const float* __restrict__ b,
                            long long total, int F) {
    long long idx = (long long)blockIdx.x * blockDim.x + threadIdx.x;
    if (idx >= total) return;
    x[idx] = fmaxf(x[idx] + b[idx % F], 0.f);
}

// ---------------- gate output (HID -> 1) ------------------------------------
__global__ void gate_out_k(const float* __restrict__ hidden,
                           const float* __restrict__ Wg2,
                           const float* __restrict__ bg2,
                           float* __restrict__ gate, int Nn) {
    int n = blockIdx.x * blockDim.x + threadIdx.x;
    if (n >= Nn) return;
    const float* hr = hidden + (size_t)n * 64;
    float s = bg2[0];
    for (int i = 0; i < 64; ++i) s += hr[i] * Wg2[i];
    gate[n] = s;
}

__global__ void gmax_k(const float* __restrict__ gate, float* __restrict__ red, int Nn) {
    int n = blockIdx.x * blockDim.x + threadIdx.x;
    if (n < Nn) atomicMaxF(&red[0], gate[n]);
}

__global__ void gsum_k(const float* __restrict__ gate, float* __restrict__ red, int Nn) {
    int n = blockIdx.x * blockDim.x + threadIdx.x;
    if (n < Nn) atomicAdd(&red[1], expf(gate[n] - red[0]));
}

__global__ void gaccum_k(const float* __restrict__ gate,
                         const float* __restrict__ x2,
                         const float* __restrict__ red,
                         float* __restrict__ accum, int Nn) {
    int idx = blockIdx.x * blockDim.x + threadIdx.x;
    if (idx >= Nn * 64) return;
    int n = idx >> 6, c = idx & 63;
    float w = expf(gate[n] - red[0]);
    atomicAdd(&accum[c], w * x2[(size_t)n * 64 + c]);
}

__global__ void finalize_k(const float* __restrict__ accum,
                           const float* __restrict__ red,
                           float* __restrict__ out) {
    int c = threadIdx.x;
    if (c < 64) out[c] = accum[c] / red[1];
}

// ---------------------------------------------------------------------------
extern "C" void kernel_launch(void* const* d_in, const int* in_sizes, int n_in,
                              void* d_out, int out_size, void* d_ws, size_t ws_size,
                              hipStream_t stream) {
    const float*     node   = (const float*)d_in[0];
    const long long* ei     = (const long long*)d_in[1];   // int64 [2,E]
    const float*     W1     = (const float*)d_in[2];
    const float*     a_src1 = (const float*)d_in[3];
    const float*     a_dst1 = (const float*)d_in[4];
    const float*     b1     = (const float*)d_in[5];
    const float*     W2     = (const float*)d_in[6];
    const float*     a_src2 = (const float*)d_in[7];
    const float*     a_dst2 = (const float*)d_in[8];
    const float*     b2     = (const float*)d_in[9];
    const float*     Wg1    = (const float*)d_in[10];
    const float*     bg1    = (const float*)d_in[11];
    const float*     Wg2    = (const float*)d_in[12];
    const float*     bg2    = (const float*)d_in[13];

    const int EMB = 128, HID = 64, H1 = 4;
    const int Nn  = in_sizes[0] / EMB;
    const int E   = in_sizes[1] / 2;
    const int Esl = E + Nn;
    const long long* srcA = ei;
    const long long* dstA = ei + E;

    // ---------------- workspace layout (floats) ----------------
    float* ws = (float*)d_ws;
    size_t off = 0;
    float* xp1   = ws + off; off += (size_t)Nn * H1 * HID;   // [N,4,64]
    float* x1    = ws + off; off += (size_t)Nn * H1 * HID;   // agg -> elu in place
    float* asrc1 = ws + off; off += (size_t)Nn * H1;
    float* adst1 = ws + off; off += (size_t)Nn * H1;
    float* m1    = ws + off; off += (size_t)Nn * H1;
    float* den1  = ws + off; off += (size_t)Nn * H1;
    float* xp2   = ws + off; off += (size_t)Nn * HID;
    float* x2    = ws + off; off += (size_t)Nn * HID;
    float* asrc2 = ws + off; off += (size_t)Nn;
    float* adst2 = ws + off; off += (size_t)Nn;
    float* m2    = ws + off; off += (size_t)Nn;
    float* den2  = ws + off; off += (size_t)Nn;
    float* hid   = ws + off; off += (size_t)Nn * HID;
    float* gate  = ws + off; off += (size_t)Nn;
    float* red   = ws + off; off += 2;                        // gmax, gsum
    float* accum = ws + off; off += HID;

    auto fill = [&](float* p, long long n, float v) {
        fill_k<<<(unsigned)((n + 255) / 256), 256, 0, stream>>>(p, n, v);
    };
    const float NEG_INF = -INFINITY;
    const int rowTiles = (Nn + 15) / 16;
    const dim3 gblk(128);

    // ================= Layer 1 =================
    fill(x1,   (long long)Nn * H1 * HID, 0.f);
    fill(m1,   (long long)Nn * H1, NEG_INF);
    fill(den1, (long long)Nn * H1, 0.f);

    gemm_wmma_f32<<<dim3((rowTiles + 3) / 4, (H1 * HID) / 16), gblk, 0, stream>>>(
        node, W1, xp1, Nn, EMB, H1 * HID);
    attn_coef_k<<<(Nn * H1 + 255) / 256, 256, 0, stream>>>(
        xp1, a_src1, a_dst1, asrc1, adst1, Nn, H1, HID);

    int egrid = (Esl * H1 + 255) / 256;
    edge_max_k<<<egrid, 256, 0, stream>>>(srcA, dstA, E, Esl, asrc1, adst1, m1, H1);
    edge_sum_k<<<egrid, 256, 0, stream>>>(srcA, dstA, E, Esl, asrc1, adst1, m1, den1, H1);
    edge_scatter_k<<<egrid, 256, 0, stream>>>(srcA, dstA, E, Esl,
        asrc1, adst1, m1, den1, xp1, x1, H1, HID);
    elu_bias_k<<<(unsigned)(((long long)Nn * H1 * HID + 255) / 256), 256, 0, stream>>>(
        x1, b1, (long long)Nn * H1 * HID, H1 * HID);

    // ================= Layer 2 =================
    fill(x2,   (long long)Nn * HID, 0.f);
    fill(m2,   (long long)Nn, NEG_INF);
    fill(den2, (long long)Nn, 0.f);

    gemm_wmma_f32<<<dim3((rowTiles + 3) / 4, HID / 16), gblk, 0, stream>>>(
        x1, W2, xp2, Nn, H1 * HID, HID);
    attn_coef_k<<<(Nn + 255) / 256, 256, 0, stream>>>(
        xp2, a_src2, a_dst2, asrc2, adst2, Nn, 1, HID);

    int egrid2 = (Esl + 255) / 256;
    edge_max_k<<<egrid2, 256, 0, stream>>>(srcA, dstA, E, Esl, asrc2, adst2, m2, 1);
    edge_sum_k<<<egrid2, 256, 0, stream>>>(srcA, dstA, E, Esl, asrc2, adst2, m2, den2, 1);
    edge_scatter_k<<<egrid2, 256, 0, stream>>>(srcA, dstA, E, Esl,
        asrc2, adst2, m2, den2, xp2, x2, 1, HID);
    elu_bias_k<<<(unsigned)(((long long)Nn * HID + 255) / 256), 256, 0, stream>>>(
        x2, b2, (long long)Nn * HID, HID);

    // ================= Attentional aggregation =================
    fill(red,     1, NEG_INF);      // gmax
    fill(red + 1, 1, 0.f);          // gsum
    fill(accum, HID, 0.f);

    // gate hidden layer: hid = relu(x2 @ Wg1 + bg1)  -- WMMA GEMM + bias/relu
    gemm_wmma_f32<<<dim3((rowTiles + 3) / 4, HID / 16), gblk, 0, stream>>>(
        x2, Wg1, hid, Nn, HID, HID);
    relu_bias_k<<<(unsigned)(((long long)Nn * HID + 255) / 256), 256, 0, stream>>>(
        hid, bg1, (long long)Nn * HID, HID);

    gate_out_k<<<(Nn + 255) / 256, 256, 0, stream>>>(hid, Wg2, bg2, gate, Nn);
    gmax_k<<<(Nn + 255) / 256, 256, 0, stream>>>(gate, red, Nn);
    gsum_k<<<(Nn + 255) / 256, 256, 0, stream>>>(gate, red, Nn);
    gaccum_k<<<(Nn * 64 + 255) / 256, 256, 0, stream>>>(gate, x2, red, accum, Nn);
    finalize_k<<<1, 64, 0, stream>>>(accum, red, (float*)d_out);
}